// SRNN_64579128262671
// MI455X (gfx1250) — compile-verified
//
#include <hip/hip_runtime.h>
#include <hip/hip_bf16.h>
#include <math.h>

// ---------------------------------------------------------------------------
// SRNN forward for MI455X (gfx1250, wave32).
// All dense GEMMs (attention S/PV, temporal embedding, LSTM gates) run on
// V_WMMA_F32_16X16X4_F32 (the fp32 tensor path; fp32 is required because
// attention scores reach ~2.6e5 after the x4095 temperature scaling).
// ---------------------------------------------------------------------------

#define TT   8
#define NN   4096
#define DD   64
#define EMB  64
#define TEMB 64
#define OUT  5
#define XK   192        // EMB + TEMB + D (concat input to LSTM gates)
#define G4D  256        // 4*D gate columns

#define BM   32         // query rows per attention block
#define BN   64         // key cols per tile
#define ATHREADS 128    // 4 waves
#define NVREG (BN * DD / (4 * ATHREADS))   // float4 regs per thread per key tile (=8)

#define CROWS 32        // rows per cell block

typedef float v2f __attribute__((ext_vector_type(2)));
typedef float v8f __attribute__((ext_vector_type(8)));

__device__ __forceinline__ float sigmoidf_(float x) {
    return 1.0f / (1.0f + __expf(-x));
}

// ---------------------------------------------------------------------------
// One-time weight prep: WcatT[k][n] = (k<128 ? W_ih[n][k] : W_hh[n][k-128])
// so gate-GEMM B reads are coalesced; bcomb = b_ih + b_hh.
// ---------------------------------------------------------------------------
__global__ void srnn_prep_kernel(const float* __restrict__ W_ih,
                                 const float* __restrict__ W_hh,
                                 const float* __restrict__ b_ih,
                                 const float* __restrict__ b_hh,
                                 float* __restrict__ WcatT,
                                 float* __restrict__ bcomb) {
    int idx = blockIdx.x * 256 + threadIdx.x;
    if (idx < XK * G4D) {
        int k = idx / G4D, n = idx % G4D;
        WcatT[idx] = (k < 128) ? W_ih[(size_t)n * 128 + k]
                               : W_hh[(size_t)n * DD + (k - 128)];
    }
    if (idx < G4D) bcomb[idx] = b_ih[idx] + b_hh[idx];
}

// ---------------------------------------------------------------------------
// Flash attention: h_spatial = softmax((h h^T) * temp, diag=-inf) @ h
// Block = 32 query rows, 4 waves; each wave owns 2 of the 8 (2x4) 16x16 tiles
// of both S and O. O stays in registers; alpha-rescale fused into PV C input.
// ---------------------------------------------------------------------------
__launch_bounds__(ATHREADS)
__global__ void srnn_attn_kernel(const float* __restrict__ h,
                                 float* __restrict__ h_spatial) {
    __shared__ float hq[BM][DD + 1];
    __shared__ float hk[2][BN][DD + 1];    // double-buffered key tile
    __shared__ float sc[BM][BN + 1];       // scores -> probabilities
    __shared__ float mrow[BM], lrow[BM], arow[BM];
    __shared__ float pmax[BM][4], psum[BM][4];

    const int tid   = threadIdx.x;
    const int lane  = tid & 31;
    const int wave  = tid >> 5;            // 0..3
    const int half  = lane >> 4;           // lane groups 0-15 / 16-31
    const int l16   = lane & 15;
    const int qbase = blockIdx.x * BM;
    const float temp = (float)(NN - 1);

    const int m0a = (((wave * 2 + 0) >> 2)) * 16, n0a = ((wave * 2 + 0) & 3) * 16;
    const int m0b = (((wave * 2 + 1) >> 2)) * 16, n0b = ((wave * 2 + 1) & 3) * 16;

    v8f oA = {}, oB = {};                  // register-resident O accumulators

    {
        const float4* src = (const float4*)(h + (size_t)qbase * DD);
        for (int i = tid; i < BM * DD / 4; i += ATHREADS) {
            float4 v = src[i];
            int r = (i * 4) >> 6, c = (i * 4) & 63;
            hq[r][c] = v.x; hq[r][c + 1] = v.y; hq[r][c + 2] = v.z; hq[r][c + 3] = v.w;
        }
    }
    if (tid < BM) { mrow[tid] = -INFINITY; lrow[tid] = 0.0f; }
    {
        const float4* src = (const float4*)h;
        #pragma unroll
        for (int j = 0; j < NVREG; ++j) {
            float4 v = src[tid + j * ATHREADS];
            int idx = (tid + j * ATHREADS) * 4;
            int r = idx >> 6, c = idx & 63;
            hk[0][r][c] = v.x; hk[0][r][c + 1] = v.y; hk[0][r][c + 2] = v.z; hk[0][r][c + 3] = v.w;
        }
    }
    __syncthreads();

    const int NKB = NN / BN;               // 64 key tiles
    for (int kb = 0; kb < NKB; ++kb) {
        const int cur = kb & 1, nxt = cur ^ 1;

        float4 nv[NVREG];
        const bool have_next = (kb + 1 < NKB);
        if (have_next) {
            const float4* src = (const float4*)(h + (size_t)(kb + 1) * BN * DD);
            #pragma unroll
            for (int j = 0; j < NVREG; ++j) nv[j] = src[tid + j * ATHREADS];
        }
        if (kb + 2 < NKB)   // global_prefetch_b8 toward L2
            __builtin_prefetch(&h[(size_t)(kb + 2) * BN * DD + tid * 32], 0, 0);

        // ---- S = (hq @ hk^T) * temp, diag masked ----
        auto s_tile = [&](int m0, int n0) {
            v8f cacc = {};
            #pragma unroll
            for (int kk = 0; kk < DD; kk += 4) {
                const int k0 = kk + half * 2;
                v2f a, b;
                a.x = hq[m0 + l16][k0];
                a.y = hq[m0 + l16][k0 + 1];
                b.x = hk[cur][n0 + l16][k0];
                b.y = hk[cur][n0 + l16][k0 + 1];
                cacc = __builtin_amdgcn_wmma_f32_16x16x4_f32(
                    false, a, false, b, (short)0, cacc, false, false);
            }
            #pragma unroll
            for (int j = 0; j < 8; ++j) {
                const int r = m0 + half * 8 + j;
                const int c = n0 + l16;
                float v = cacc[j] * temp;
                if (qbase + r == kb * BN + c) v = -INFINITY;
                sc[r][c] = v;
            }
        };
        s_tile(m0a, n0a);
        s_tile(m0b, n0b);
        __syncthreads();

        // ---- online softmax, 4 threads per row ----
        const int sr = tid >> 2, sq = tid & 3;
        {
            float rmax = -INFINITY;
            #pragma unroll
            for (int c = 0; c < 16; ++c) rmax = fmaxf(rmax, sc[sr][sq * 16 + c]);
            pmax[sr][sq] = rmax;
        }
        __syncthreads();
        if (tid < BM) {
            const float mold = mrow[tid];
            float mnew = fmaxf(fmaxf(pmax[tid][0], pmax[tid][1]),
                               fmaxf(pmax[tid][2], pmax[tid][3]));
            mnew = fmaxf(mold, mnew);
            mrow[tid] = mnew;
            arow[tid] = __expf(mold - mnew);
        }
        __syncthreads();
        {
            const float mnew = mrow[sr];
            float s = 0.0f;
            #pragma unroll
            for (int c = 0; c < 16; ++c) {
                float p = __expf(sc[sr][sq * 16 + c] - mnew);
                sc[sr][sq * 16 + c] = p;
                s += p;
            }
            psum[sr][sq] = s;
        }
        __syncthreads();
        if (tid < BM) {
            lrow[tid] = lrow[tid] * arow[tid]
                      + psum[tid][0] + psum[tid][1] + psum[tid][2] + psum[tid][3];
        }

        // ---- O = O*alpha + P @ hk ----
        auto pv_tile = [&](int m0, int n0, v8f o) {
            #pragma unroll
            for (int j = 0; j < 8; ++j) o[j] *= arow[m0 + half * 8 + j];
            #pragma unroll
            for (int kk = 0; kk < BN; kk += 4) {
                const int k0 = kk + half * 2;
                v2f a, b;
                a.x = sc[m0 + l16][k0];
                a.y = sc[m0 + l16][k0 + 1];
                b.x = hk[cur][k0][n0 + l16];
                b.y = hk[cur][k0 + 1][n0 + l16];
                o = __builtin_amdgcn_wmma_f32_16x16x4_f32(
                    false, a, false, b, (short)0, o, false, false);
            }
            return o;
        };
        oA = pv_tile(m0a, n0a, oA);
        oB = pv_tile(m0b, n0b, oB);

        if (have_next) {
            #pragma unroll
            for (int j = 0; j < NVREG; ++j) {
                int idx = (tid + j * ATHREADS) * 4;
                int r = idx >> 6, c = idx & 63;
                hk[nxt][r][c]     = nv[j].x;
                hk[nxt][r][c + 1] = nv[j].y;
                hk[nxt][r][c + 2] = nv[j].z;
                hk[nxt][r][c + 3] = nv[j].w;
            }
        }
        __syncthreads();
    }

    #pragma unroll
    for (int j = 0; j < 8; ++j) {
        const int rA = m0a + half * 8 + j;
        const int rB = m0b + half * 8 + j;
        h_spatial[(size_t)(qbase + rA) * DD + n0a + l16] = oA[j] / lrow[rA];
        h_spatial[(size_t)(qbase + rB) * DD + n0b + l16] = oB[j] / lrow[rB];
    }
}

// ---------------------------------------------------------------------------
// Fused cell: enc (VALU, K=2), ht and LSTM gates on WMMA, elementwise LSTM,
// output projection. 32 rows per block (128 blocks), 256 threads = 8 waves.
// ---------------------------------------------------------------------------
__launch_bounds__(256)
__global__ void srnn_cell_kernel(const float* __restrict__ nodes_t,
                                 const float* __restrict__ h,
                                 const float* __restrict__ c,
                                 const float* __restrict__ h_spatial,
                                 const float* __restrict__ W_enc,
                                 const float* __restrict__ b_enc,
                                 const float* __restrict__ W_t,
                                 const float* __restrict__ b_t,
                                 const float* __restrict__ WcatT,   // [192][256]
                                 const float* __restrict__ bcomb,   // [256]
                                 const float* __restrict__ W_out,
                                 const float* __restrict__ b_out,
                                 float* __restrict__ h_next,
                                 float* __restrict__ c_next,
                                 float* __restrict__ out_t) {
    __shared__ float xf[CROWS][XK + 1];      // [enc | ht | h] per row
    __shared__ float hs[CROWS][DD + 1];      // staged h_spatial
    __shared__ float gsh[CROWS][G4D + 1];    // gate pre-activations
    __shared__ float hn[CROWS][DD + 1];      // staged h_new

    const int tid   = threadIdx.x;
    const int lane  = tid & 31;
    const int wave  = tid >> 5;              // 0..7
    const int half  = lane >> 4;
    const int l16   = lane & 15;
    const int rbase = blockIdx.x * CROWS;

    // stage h_spatial and the h-part of the concat input; compute enc (K=2)
    for (int i = tid; i < CROWS * DD; i += 256) {
        int r = i >> 6, j = i & 63;
        hs[r][j]          = h_spatial[(size_t)(rbase + r) * DD + j];
        xf[r][2 * DD + j] = h[(size_t)(rbase + r) * DD + j];
        float p0 = nodes_t[(size_t)(rbase + r) * 2 + 0];
        float p1 = nodes_t[(size_t)(rbase + r) * 2 + 1];
        float v = p0 * W_enc[j] + p1 * W_enc[EMB + j] + b_enc[j];
        xf[r][j] = fmaxf(v, 0.0f);
    }
    __syncthreads();

    // ---- ht = relu(hs @ W_t + b_t): 8 tiles (2x4), one per wave ----
    {
        const int m0 = (wave >> 2) * 16, n0 = (wave & 3) * 16;
        v8f acc;
        const float bt = b_t[n0 + l16];
        #pragma unroll
        for (int j = 0; j < 8; ++j) acc[j] = bt;
        #pragma unroll
        for (int kk = 0; kk < DD; kk += 4) {
            const int k0 = kk + half * 2;
            v2f a, b;
            a.x = hs[m0 + l16][k0];
            a.y = hs[m0 + l16][k0 + 1];
            b.x = W_t[(size_t)k0 * TEMB + n0 + l16];        // W_t is K-major
            b.y = W_t[(size_t)(k0 + 1) * TEMB + n0 + l16];
            acc = __builtin_amdgcn_wmma_f32_16x16x4_f32(
                false, a, false, b, (short)0, acc, false, false);
        }
        #pragma unroll
        for (int j = 0; j < 8; ++j)
            xf[m0 + half * 8 + j][EMB + n0 + l16] = fmaxf(acc[j], 0.0f);
    }
    __syncthreads();

    // ---- gates = xf @ WcatT + bcomb: 2x16 tiles, 4 per wave (shared A) ----
    {
        const int m0 = (wave >> 2) * 16;            // row tile (const per wave)
        const int cg = (wave & 3) * 4;              // first of 4 col tiles
        v8f g0, g1, g2, g3;
        #pragma unroll
        for (int j = 0; j < 8; ++j) {
            g0[j] = bcomb[(cg + 0) * 16 + l16];
            g1[j] = bcomb[(cg + 1) * 16 + l16];
            g2[j] = bcomb[(cg + 2) * 16 + l16];
            g3[j] = bcomb[(cg + 3) * 16 + l16];
        }
        for (int kk = 0; kk < XK; kk += 4) {
            const int k0 = kk + half * 2;
            v2f a;
            a.x = xf[m0 + l16][k0];
            a.y = xf[m0 + l16][k0 + 1];
            const float* w0 = WcatT + (size_t)k0 * G4D;
            const float* w1 = WcatT + (size_t)(k0 + 1) * G4D;
            v2f b0, b1, b2, b3;
            b0.x = w0[(cg + 0) * 16 + l16]; b0.y = w1[(cg + 0) * 16 + l16];
            b1.x = w0[(cg + 1) * 16 + l16]; b1.y = w1[(cg + 1) * 16 + l16];
            b2.x = w0[(cg + 2) * 16 + l16]; b2.y = w1[(cg + 2) * 16 + l16];
            b3.x = w0[(cg + 3) * 16 + l16]; b3.y = w1[(cg + 3) * 16 + l16];
            g0 = __builtin_amdgcn_wmma_f32_16x16x4_f32(false, a, false, b0, (short)0, g0, false, false);
            g1 = __builtin_amdgcn_wmma_f32_16x16x4_f32(false, a, false, b1, (short)0, g1, false, false);
            g2 = __builtin_amdgcn_wmma_f32_16x16x4_f32(false, a, false, b2, (short)0, g2, false, false);
            g3 = __builtin_amdgcn_wmma_f32_16x16x4_f32(false, a, false, b3, (short)0, g3, false, false);
        }
        #pragma unroll
        for (int j = 0; j < 8; ++j) {
            const int r = m0 + half * 8 + j;
            gsh[r][(cg + 0) * 16 + l16] = g0[j];
            gsh[r][(cg + 1) * 16 + l16] = g1[j];
            gsh[r][(cg + 2) * 16 + l16] = g2[j];
            gsh[r][(cg + 3) * 16 + l16] = g3[j];
        }
    }
    __syncthreads();

    // ---- elementwise LSTM (gate order i,f,g,o) ----
    for (int it = tid; it < CROWS * DD; it += 256) {
        const int r = it >> 6, d = it & 63;
        const float gi = gsh[r][0 * DD + d];
        const float gf = gsh[r][1 * DD + d];
        const float gg = gsh[r][2 * DD + d];
        const float go = gsh[r][3 * DD + d];
        const float cold = c[(size_t)(rbase + r) * DD + d];
        const float cnew = sigmoidf_(gf) * cold + sigmoidf_(gi) * tanhf(gg);
        const float hnew = sigmoidf_(go) * tanhf(cnew);
        c_next[(size_t)(rbase + r) * DD + d] = cnew;
        h_next[(size_t)(rbase + r) * DD + d] = hnew;
        hn[r][d] = hnew;
    }
    __syncthreads();

    // ---- out = h_new @ W_out + b_out (OUT=5, trivial) ----
    for (int i = tid; i < CROWS * OUT; i += 256) {
        int r = i / OUT, o = i % OUT;
        float acc = b_out[o];
        #pragma unroll 8
        for (int d = 0; d < DD; ++d) acc += hn[r][d] * W_out[d * OUT + o];
        out_t[(size_t)(rbase + r) * OUT + o] = acc;
    }
}

// ---------------------------------------------------------------------------
// Host launcher
// ---------------------------------------------------------------------------
extern "C" void kernel_launch(void* const* d_in, const int* in_sizes, int n_in,
                              void* d_out, int out_size, void* d_ws, size_t ws_size,
                              hipStream_t stream) {
    const float* nodes = (const float*)d_in[0];
    const float* h0    = (const float*)d_in[1];
    const float* c0    = (const float*)d_in[2];
    const float* W_enc = (const float*)d_in[3];
    const float* b_enc = (const float*)d_in[4];
    const float* W_t   = (const float*)d_in[5];
    const float* b_t   = (const float*)d_in[6];
    const float* W_ih  = (const float*)d_in[7];
    const float* W_hh  = (const float*)d_in[8];
    const float* b_ih  = (const float*)d_in[9];
    const float* b_hh  = (const float*)d_in[10];
    const float* W_out = (const float*)d_in[11];
    const float* b_out = (const float*)d_in[12];

    const size_t ND = (size_t)NN * DD;
    float* ws    = (float*)d_ws;
    float* hc    = ws + 0 * ND;   // current h
    float* hn    = ws + 1 * ND;   // next h
    float* cc    = ws + 2 * ND;   // current c
    float* cn    = ws + 3 * ND;   // next c
    float* hsp   = ws + 4 * ND;   // h_spatial scratch
    float* WcatT = ws + 5 * ND;   // [192][256] transposed gate weights
    float* bcomb = WcatT + (size_t)XK * G4D;   // [256]

    float* outs = (float*)d_out;                 // [T, N, OUT]
    float* hT   = outs + (size_t)TT * NN * OUT;  // [N, D]
    float* cT   = hT + ND;                       // [N, D]

    hipMemcpyAsync(hc, h0, ND * sizeof(float), hipMemcpyDeviceToDevice, stream);
    hipMemcpyAsync(cc, c0, ND * sizeof(float), hipMemcpyDeviceToDevice, stream);

    srnn_prep_kernel<<<(XK * G4D + 255) / 256, 256, 0, stream>>>(
        W_ih, W_hh, b_ih, b_hh, WcatT, bcomb);

    for (int t = 0; t < TT; ++t) {
        srnn_attn_kernel<<<NN / BM, ATHREADS, 0, stream>>>(hc, hsp);
        srnn_cell_kernel<<<NN / CROWS, 256, 0, stream>>>(
            nodes + (size_t)t * NN * 2, hc, cc, hsp,
            W_enc, b_enc, W_t, b_t, WcatT, bcomb, W_out, b_out,
            hn, cn, outs + (size_t)t * NN * OUT);
        float* tmp;
        tmp = hc; hc = hn; hn = tmp;
        tmp = cc; cc = cn; cn = tmp;
    }

    hipMemcpyAsync(hT, hc, ND * sizeof(float), hipMemcpyDeviceToDevice, stream);
    hipMemcpyAsync(cT, cc, ND * sizeof(float), hipMemcpyDeviceToDevice, stream);
}